// TopkActivation_52192442581571
// MI455X (gfx1250) — compile-verified
//
#include <hip/hip_runtime.h>
#include <hip/hip_bf16.h>

// Top-k (k=64) per row of an 8192x32768 fp32 matrix, scattered back into a
// zero tensor. Memory-bound: 1 GiB in + 1 GiB out => ~92us floor @ 23.3 TB/s.
// One workgroup per row. The 128KB row is staged in LDS with CDNA5 async
// global->LDS copies (ASYNCcnt); pass-0 of an exact 4-pass radix select is
// fused with the DMA (per-wave s_wait_asynccnt pipelining, legal because
// ASYNCcnt completes in order and each thread reads only LDS its own wave's
// async ops wrote). Threshold-bin selection runs in a single wave with
// __shfl_down suffix scans (no barriered tree scan). Output uses
// non-temporal b128 stores (streamed, never re-read).

typedef float v4f __attribute__((ext_vector_type(4)));

#define THREADS 1024
#define ROW_LEN 32768
#define VPT 8 // ROW_LEN / 4 / THREADS  (float4 chunks per thread)

// Order-preserving map: ascending unsigned key == ascending float.
__device__ __forceinline__ unsigned f2key(unsigned u) {
    return u ^ ((unsigned)((int)u >> 31) | 0x80000000u);
}

__global__ __launch_bounds__(THREADS)
void topk_scatter_kernel(const float* __restrict__ x,
                         const int* __restrict__ kptr,
                         float* __restrict__ out) {
    extern __shared__ unsigned char smem[];              // 131072 B row staging
    __shared__ __align__(16) unsigned hist8[256][8];     // 8-way privatized bins
    __shared__ unsigned sh_prefix, sh_mask, sh_krem, sh_eq;

    const unsigned tid = threadIdx.x;
    const size_t   row = blockIdx.x;
    const unsigned k   = (unsigned)*kptr;

    // Low 32 bits of the flat address of an LDS object == workgroup-relative
    // LDS byte offset (ISA 10.2), which the async op's VGPR expects.
    const unsigned sbase = (unsigned)(size_t)(void*)smem;
    const unsigned long long gbase =
        (unsigned long long)(const void*)(x + row * (size_t)ROW_LEN);

    // --- Kick off all async staging (HBM -> LDS) immediately ---
#pragma unroll
    for (int j = 0; j < VPT; ++j) {
        const unsigned byteOff = (tid + j * THREADS) * 16u;
        asm volatile("global_load_async_to_lds_b128 %0, %1, off"
                     :: "v"(sbase + byteOff),
                        "v"(gbase + (unsigned long long)byteOff)
                     : "memory");
    }

    // Zero histogram + scalars while the DMA is in flight.
    for (unsigned i = tid; i < 256u * 8u; i += THREADS)
        ((unsigned*)hist8)[i] = 0;
    if (tid == 0) sh_eq = 0;
    __syncthreads(); // barrier is legal with outstanding ASYNCcnt

    const uint4*   sbuf = (const uint4*)smem;
    const unsigned copy = tid >> 7; // 4 waves share one histogram copy

    // --- Pass 0 (byte [31:24]) fused with per-wave async completion.
    // ASYNCcnt decrements in order; waiting <= (7-j) guarantees chunk j of
    // this wave has landed in LDS, and each thread reads only its own chunks.
#define P0_STEP(J, W)                                                   \
    do {                                                                \
        asm volatile("s_wait_asynccnt " W ::: "memory");                \
        const uint4 w = sbuf[tid + (J) * THREADS];                      \
        atomicAdd(&hist8[f2key(w.x) >> 24][copy], 1u);                  \
        atomicAdd(&hist8[f2key(w.y) >> 24][copy], 1u);                  \
        atomicAdd(&hist8[f2key(w.z) >> 24][copy], 1u);                  \
        atomicAdd(&hist8[f2key(w.w) >> 24][copy], 1u);                  \
    } while (0)
    P0_STEP(0, "7"); P0_STEP(1, "6"); P0_STEP(2, "5"); P0_STEP(3, "4");
    P0_STEP(4, "3"); P0_STEP(5, "2"); P0_STEP(6, "1"); P0_STEP(7, "0");
#undef P0_STEP
    __syncthreads();

    unsigned prefix = 0, pmask = 0, krem = k;

    // --- Exact radix-select over byte planes [31:24] .. [7:0] ---
    for (int pass = 0; pass < 4; ++pass) {
        const int shift = 24 - 8 * pass;

        // Threshold-bin selection entirely inside wave 0 (no barriered scan).
        if (tid < 32) {
            const unsigned lane = tid;
            unsigned c[8], suf[9];
#pragma unroll
            for (int i = 0; i < 8; ++i) { // combine the 8 privatized copies
                const unsigned bin = lane * 8u + (unsigned)i;
                const uint4 a = *(const uint4*)&hist8[bin][0];
                const uint4 b = *(const uint4*)&hist8[bin][4];
                c[i] = a.x + a.y + a.z + a.w + b.x + b.y + b.z + b.w;
            }
            suf[8] = 0; // within-lane suffix sums (bins high -> low)
#pragma unroll
            for (int i = 7; i >= 0; --i) suf[i] = suf[i + 1] + c[i];
            unsigned s = suf[0];
            unsigned S = s; // inclusive cross-lane suffix via shuffles
#pragma unroll
            for (int d = 1; d < 32; d <<= 1) {
                const unsigned o = __shfl_down(S, (unsigned)d);
                if (lane + (unsigned)d < 32u) S += o;
            }
            const unsigned E = S - s; // count in bins of lanes above this one
#pragma unroll
            for (int i = 0; i < 8; ++i) {
                const unsigned cur = suf[i] + E;     // suffix[bin]
                const unsigned nxt = suf[i + 1] + E; // suffix[bin+1]
                if (cur >= krem && nxt < krem) {
                    sh_prefix = prefix | ((lane * 8u + (unsigned)i) << shift);
                    sh_mask   = pmask | (0xFFu << shift);
                    sh_krem   = krem - nxt;
                }
            }
        }
        __syncthreads();
        prefix = sh_prefix;
        pmask  = sh_mask;
        krem   = sh_krem;

        if (pass < 3) { // build next byte-plane histogram from LDS
            for (unsigned i = tid; i < 256u * 8u; i += THREADS)
                ((unsigned*)hist8)[i] = 0;
            __syncthreads();
            const int nshift = shift - 8;
#pragma unroll
            for (int j = 0; j < VPT; ++j) {
                const uint4 w = sbuf[tid + j * THREADS];
                const unsigned kk[4] = {f2key(w.x), f2key(w.y),
                                        f2key(w.z), f2key(w.w)};
#pragma unroll
                for (int cc = 0; cc < 4; ++cc) {
                    if ((kk[cc] & pmask) == prefix)
                        atomicAdd(&hist8[(kk[cc] >> nshift) & 0xFFu][copy], 1u);
                }
            }
            __syncthreads();
        }
    }

    // --- Emit: key > T kept; exactly krem of the key == T ties kept ---
    const unsigned T = prefix;
    float* orow = out + row * (size_t)ROW_LEN;

#pragma unroll
    for (int j = 0; j < VPT; ++j) {
        const unsigned idx = tid + j * THREADS;
        const uint4 w = sbuf[idx];
        v4f o;
#pragma unroll
        for (int c = 0; c < 4; ++c) {
            const unsigned u   = (&w.x)[c];
            const unsigned key = f2key(u);
            float v = 0.0f;
            if (key > T) {
                v = __uint_as_float(u);
            } else if (key == T) {
                if (atomicAdd(&sh_eq, 1u) < krem) v = __uint_as_float(u);
            }
            o[c] = v;
        }
        __builtin_nontemporal_store(o, (v4f*)orow + idx);
    }
}

extern "C" void kernel_launch(void* const* d_in, const int* in_sizes, int n_in,
                              void* d_out, int out_size, void* d_ws, size_t ws_size,
                              hipStream_t stream) {
    const float* x   = (const float*)d_in[0];
    const int*   kp  = (const int*)d_in[1];
    float*       out = (float*)d_out;

    const int B = in_sizes[0] / ROW_LEN; // 8192 rows
    hipLaunchKernelGGL(topk_scatter_kernel, dim3(B), dim3(THREADS),
                       ROW_LEN * sizeof(float), stream, x, kp, out);
}